// WeightedCCEDiceLossWithSoftmax_18915035971718
// MI455X (gfx1250) — compile-verified
//
#include <hip/hip_runtime.h>

typedef __attribute__((ext_vector_type(2))) float v2f;
typedef __attribute__((ext_vector_type(8))) float v8f;

#define EPS 1e-8f

// ---------------------------------------------------------------------------
// ws layout: ws[0..63] = confusion matrix (row-major, gt x pred), ws[64] = CE sum
// ---------------------------------------------------------------------------
__global__ void zero_ws_kernel(float* __restrict__ ws) {
  if (threadIdx.x < 65) ws[threadIdx.x] = 0.0f;
}

// ---------------------------------------------------------------------------
// Main streaming pass: memory-bound, 2x global_load_b128 per operand row.
// ---------------------------------------------------------------------------
__global__ __launch_bounds__(256) void cce_dice_main(
    const float* __restrict__ pred, const float* __restrict__ gt,
    const float* __restrict__ wts, float* __restrict__ ws, int B) {
  __shared__ float cm[64];
  __shared__ float wsum_lds[8];
  for (int i = threadIdx.x; i < 64; i += 256) cm[i] = 0.0f;
  __syncthreads();

  float4 wlo = *(const float4*)(wts);
  float4 whi = *(const float4*)(wts + 4);
  float w[8] = {wlo.x, wlo.y, wlo.z, wlo.w, whi.x, whi.y, whi.z, whi.w};

  float ce_sum = 0.0f;
  int tid = blockIdx.x * 256 + threadIdx.x;
  int stride = gridDim.x * 256;
  for (int i = tid; i < B; i += stride) {
    const float4* pr = (const float4*)(pred + (size_t)i * 8);
    float4 x0 = pr[0];
    float4 x1 = pr[1];
    const float4* gr = (const float4*)(gt + (size_t)i * 8);
    float4 g0 = gr[0];
    float4 g1 = gr[1];
    float x[8] = {x0.x, x0.y, x0.z, x0.w, x1.x, x1.y, x1.z, x1.w};
    float g[8] = {g0.x, g0.y, g0.z, g0.w, g1.x, g1.y, g1.z, g1.w};

    // argmax (first-max tie-break, matches jnp.argmax) + max for stable softmax
    float m = x[0];
    int pc = 0;
#pragma unroll
    for (int c = 1; c < 8; ++c) {
      if (x[c] > m) { m = x[c]; pc = c; }
    }

    float e[8], s = 0.0f;
#pragma unroll
    for (int c = 0; c < 8; ++c) {
      e[c] = __expf(x[c] - m);
      s += e[c];
    }
    float inv = __builtin_amdgcn_rcpf(s);

    // p = softmax(x); t = sum exp(p) (p in (0,1], no overflow risk);
    // one-hot dot products avoid dynamic register indexing.
    float t = 0.0f, pg = 0.0f, wg = 0.0f;
    int gc = 0;
#pragma unroll
    for (int c = 0; c < 8; ++c) {
      float p = e[c] * inv;
      t += __expf(p);
      pg = fmaf(p, g[c], pg);
      wg = fmaf(w[c], g[c], wg);
      gc = (g[c] > 0.5f) ? c : gc;
    }
    // ce = -w[gt] * log_softmax(p)[gt] = -w[gt] * (p[gt] - log(sum exp(p)))
    ce_sum += -wg * (pg - __logf(t));
    atomicAdd(&cm[gc * 8 + pc], 1.0f);  // ds_add_f32
  }

  // wave32 tree reduction of CE
#pragma unroll
  for (int off = 16; off > 0; off >>= 1)
    ce_sum += __shfl_down(ce_sum, off, 32);
  int lane = threadIdx.x & 31;
  int wv = threadIdx.x >> 5;
  if (lane == 0) wsum_lds[wv] = ce_sum;
  __syncthreads();
  if (threadIdx.x == 0) {
    float bs = 0.0f;
#pragma unroll
    for (int k = 0; k < 8; ++k) bs += wsum_lds[k];
    atomicAdd(&ws[64], bs);
  }
  // flush per-block confusion matrix (barrier above guarantees cm complete)
  for (int i = threadIdx.x; i < 64; i += 256) atomicAdd(&ws[i], cm[i]);
}

// ---------------------------------------------------------------------------
// Finalize: one wave. Row/col sums of the 8x8 confusion matrix via
// V_WMMA_F32_16X16X4_F32 (cm x ones and cm^T x ones, K split 0-3 / 4-7).
// A layout per ISA: lane (m = lane&15) holds A[m][kh], A[m][kh+1], kh = 2*(lane>=16).
// D layout: on lane j<16, acc[r] = D[r][j] = desired sum for class r.
// ---------------------------------------------------------------------------
__global__ void cce_dice_finalize(const float* __restrict__ ws,
                                  const float* __restrict__ wts,
                                  float* __restrict__ out, int B) {
  int lane = threadIdx.x;       // 32 threads, EXEC all ones for WMMA
  int m = lane & 15;
  int kh = (lane >> 4) << 1;    // 0 or 2
  v2f a1 = {0.0f, 0.0f}, a2 = {0.0f, 0.0f};
  v2f t1 = {0.0f, 0.0f}, t2 = {0.0f, 0.0f};
  if (m < 8) {
    a1[0] = ws[m * 8 + kh + 0];        // cm[m][k], k = kh..kh+1
    a1[1] = ws[m * 8 + kh + 1];
    a2[0] = ws[m * 8 + kh + 4];        // cm[m][k+4]
    a2[1] = ws[m * 8 + kh + 5];
    t1[0] = ws[(kh + 0) * 8 + m];      // cm^T
    t1[1] = ws[(kh + 1) * 8 + m];
    t2[0] = ws[(kh + 4) * 8 + m];
    t2[1] = ws[(kh + 5) * 8 + m];
  }
  v2f ones = {1.0f, 1.0f};
  v8f rs = {0, 0, 0, 0, 0, 0, 0, 0};
  rs = __builtin_amdgcn_wmma_f32_16x16x4_f32(false, a1, false, ones, (short)0, rs, false, false);
  rs = __builtin_amdgcn_wmma_f32_16x16x4_f32(false, a2, false, ones, (short)0, rs, false, false);
  v8f cs = {0, 0, 0, 0, 0, 0, 0, 0};
  cs = __builtin_amdgcn_wmma_f32_16x16x4_f32(false, t1, false, ones, (short)0, cs, false, false);
  cs = __builtin_amdgcn_wmma_f32_16x16x4_f32(false, t2, false, ones, (short)0, cs, false, false);

  if (lane == 0) {
    float rsv[8] = {rs[0], rs[1], rs[2], rs[3], rs[4], rs[5], rs[6], rs[7]};
    float csv[8] = {cs[0], cs[1], cs[2], cs[3], cs[4], cs[5], cs[6], cs[7]};
    float dsum = 0.0f;
#pragma unroll
    for (int c = 0; c < 8; ++c) {
      float tp = ws[c * 8 + c];
      float fp = csv[c] - tp;           // column sum - diag
      float fn = rsv[c] - tp;           // row sum - diag
      float dice = (tp + EPS) / (tp + fp + fn + EPS);
      dsum += (1.0f - dice) * wts[c];
    }
    float dice_loss = dsum * 0.125f;    // mean over 8 classes
    float cce = ws[64] / (float)B;      // mean over batch
    out[0] = cce + 0.5f * dice_loss;
  }
}

extern "C" void kernel_launch(void* const* d_in, const int* in_sizes, int n_in,
                              void* d_out, int out_size, void* d_ws, size_t ws_size,
                              hipStream_t stream) {
  (void)n_in; (void)out_size; (void)ws_size;
  const float* pred = (const float*)d_in[0];
  const float* gt   = (const float*)d_in[1];
  const float* wts  = (const float*)d_in[2];
  float* out = (float*)d_out;
  float* ws  = (float*)d_ws;
  int B = in_sizes[0] / 8;

  zero_ws_kernel<<<1, 128, 0, stream>>>(ws);
  cce_dice_main<<<4096, 256, 0, stream>>>(pred, gt, wts, ws, B);
  cce_dice_finalize<<<1, 32, 0, stream>>>(ws, wts, out, B);
}